// Net_730144440440
// MI455X (gfx1250) — compile-verified
//
#include <hip/hip_runtime.h>
#include <hip/hip_bf16.h>

typedef __attribute__((ext_vector_type(2))) float v2f;
typedef __attribute__((ext_vector_type(8))) float v8f;

// ---------------------------------------------------------------------------
// Graph weight preparation
// ---------------------------------------------------------------------------
__global__ void deg_scatter_kernel(const int* __restrict__ src,
                                   const float* __restrict__ attr,
                                   float* __restrict__ deg, int nE) {
    int e = blockIdx.x * blockDim.x + threadIdx.x;
    if (e >= nE) return;
    atomicAdd(&deg[src[e]], attr[e]);
}

__global__ void dis_kernel(float* __restrict__ deg, int n) {
    int i = blockIdx.x * blockDim.x + threadIdx.x;
    if (i >= n) return;
    float d = deg[i];
    deg[i] = (d > 0.f) ? rsqrtf(fmaxf(d, 1e-30f)) : 0.f;   // deg becomes dis
}

__global__ void lw_kernel(const int* __restrict__ src, const int* __restrict__ dst,
                          const float* __restrict__ attr, const float* __restrict__ dis,
                          float* __restrict__ lw, int nE) {
    int e = blockIdx.x * blockDim.x + threadIdx.x;
    if (e >= nE) return;
    lw[e] = -(dis[src[e]] * attr[e] * dis[dst[e]]);
}

// ---------------------------------------------------------------------------
// WMMA GEMM:  out[N, FOUT] = A0 @ W0 (+ A1 @ W1) + bias, optional leaky-relu.
// One wave32 owns a 16-row tile; K stepped by 4 with V_WMMA_F32_16X16X4_F32.
// ---------------------------------------------------------------------------
template<int FIN, int FOUT, bool TWO, bool LRELU>
__global__ __launch_bounds__(256)
void cheb_gemm_kernel(const float* __restrict__ A0, const float* __restrict__ A1,
                      const float* __restrict__ W0, const float* __restrict__ W1p,
                      const float* __restrict__ bias, float* __restrict__ out,
                      int n_tiles, int n_rows) {
    constexpr int NT = (FOUT + 15) / 16;
    const int wave = threadIdx.x >> 5;
    const int lane = threadIdx.x & 31;
    const int tile = blockIdx.x * 8 + wave;
    if (tile >= n_tiles) return;                 // wave-uniform: EXEC stays full

    const int arow  = min(tile * 16 + (lane & 15), n_rows - 1); // A-frag row
    const int khalf = (lane >> 4) * 2;                          // 0 or 2
    const int ncol  = lane & 15;                                // col in N-tile

    v8f acc[NT] = {};

    const float* a0p = A0 + (size_t)arow * FIN + khalf;
    #pragma unroll
    for (int k = 0; k < FIN; k += 4) {
        v2f a; a.x = a0p[k]; a.y = a0p[k + 1];
        #pragma unroll
        for (int t = 0; t < NT; ++t) {
            const int col = t * 16 + ncol;
            v2f b;
            if (FOUT >= 16 || col < FOUT) {
                b.x = W0[(size_t)(k + khalf)     * FOUT + col];
                b.y = W0[(size_t)(k + khalf + 1) * FOUT + col];
            } else { b.x = 0.f; b.y = 0.f; }
            acc[t] = __builtin_amdgcn_wmma_f32_16x16x4_f32(
                false, a, false, b, (short)0, acc[t], false, false);
        }
    }
    if (TWO) {
        const float* a1p = A1 + (size_t)arow * FIN + khalf;
        #pragma unroll
        for (int k = 0; k < FIN; k += 4) {
            v2f a; a.x = a1p[k]; a.y = a1p[k + 1];
            #pragma unroll
            for (int t = 0; t < NT; ++t) {
                const int col = t * 16 + ncol;
                v2f b;
                if (FOUT >= 16 || col < FOUT) {
                    b.x = W1p[(size_t)(k + khalf)     * FOUT + col];
                    b.y = W1p[(size_t)(k + khalf + 1) * FOUT + col];
                } else { b.x = 0.f; b.y = 0.f; }
                acc[t] = __builtin_amdgcn_wmma_f32_16x16x4_f32(
                    false, a, false, b, (short)0, acc[t], false, false);
            }
        }
    }

    // ---- epilogue: bias + activation + store -----------------------------
    // C/D layout: VGPR r -> row = tile*16 + r (+8 for lanes 16..31), col = ncol.
    const int  rbase = tile * 16 + ((lane >> 4) << 3);
    const bool full  = (tile * 16 + 16) <= n_rows;     // wave-uniform fast path

    #pragma unroll
    for (int t = 0; t < NT; ++t) {
        const int col = t * 16 + ncol;
        if (FOUT < 16 && col >= FOUT) continue;        // lane mask (narrow FOUT)
        const float bv = bias[col];
        float* op = out + (size_t)rbase * FOUT + col;
        if (full) {
            #pragma unroll
            for (int r = 0; r < 8; ++r) {
                float v = acc[t][r] + bv;
                if (LRELU) v = (v > 0.f) ? v : 0.01f * v;
                op[(size_t)r * FOUT] = v;
            }
        } else {
            #pragma unroll
            for (int r = 0; r < 8; ++r) {
                if (rbase + r >= n_rows) continue;
                float v = acc[t][r] + bv;
                if (LRELU) v = (v > 0.f) ? v : 0.01f * v;
                op[(size_t)r * FOUT] = v;
            }
        }
    }
}

// ---------------------------------------------------------------------------
// Propagation: out[dst] += lw[e] * h[src]  (thread = edge x 4 channels)
// ---------------------------------------------------------------------------
template<int F>
__global__ void prop_kernel(const float* __restrict__ h, const int* __restrict__ src,
                            const int* __restrict__ dst, const float* __restrict__ lw,
                            float* __restrict__ out, int nE) {
    constexpr int C = F / 4;
    int idx = blockIdx.x * blockDim.x + threadIdx.x;
    if (idx >= nE * C) return;
    const int e  = idx / C;
    const int c4 = (idx - e * C) * 4;
    const float w = lw[e];
    const float4 v = *(const float4*)(h + (size_t)src[e] * F + c4);
    float* op = out + (size_t)dst[e] * F + c4;
    atomicAdd(op + 0, w * v.x);
    atomicAdd(op + 1, w * v.y);
    atomicAdd(op + 2, w * v.z);
    atomicAdd(op + 3, w * v.w);
}

// ---------------------------------------------------------------------------
// Attention pool + log_softmax, single block
// ---------------------------------------------------------------------------
__global__ __launch_bounds__(512)
void finalize_kernel(const float* __restrict__ h3, const float* __restrict__ gw,
                     const float* __restrict__ gb, float* __restrict__ out, int n) {
    __shared__ float r0[512], r1[512], r2[512];
    const int tid = threadIdx.x;
    const float g0 = gw[0], g1 = gw[1], gbv = gb[0];

    float m = -3.4e38f;
    for (int i = tid; i < n; i += 512) {
        float g = h3[2 * i] * g0 + h3[2 * i + 1] * g1 + gbv;
        m = fmaxf(m, g);
    }
    r0[tid] = m; __syncthreads();
    for (int s = 256; s > 0; s >>= 1) {
        if (tid < s) r0[tid] = fmaxf(r0[tid], r0[tid + s]);
        __syncthreads();
    }
    m = r0[0]; __syncthreads();

    float S = 0.f, p0 = 0.f, p1 = 0.f;
    for (int i = tid; i < n; i += 512) {
        float a = h3[2 * i], b = h3[2 * i + 1];
        float e = expf(a * g0 + b * g1 + gbv - m);
        S += e; p0 += e * a; p1 += e * b;
    }
    r0[tid] = S; r1[tid] = p0; r2[tid] = p1; __syncthreads();
    for (int s = 256; s > 0; s >>= 1) {
        if (tid < s) { r0[tid] += r0[tid + s]; r1[tid] += r1[tid + s]; r2[tid] += r2[tid + s]; }
        __syncthreads();
    }
    if (tid == 0) {
        float pooled0 = r1[0] / r0[0];
        float pooled1 = r2[0] / r0[0];
        float mm  = fmaxf(pooled0, pooled1);
        float lse = mm + logf(expf(pooled0 - mm) + expf(pooled1 - mm));
        out[0] = pooled0 - lse;
        out[1] = pooled1 - lse;
    }
}

// ---------------------------------------------------------------------------
extern "C" void kernel_launch(void* const* d_in, const int* in_sizes, int n_in,
                              void* d_out, int out_size, void* d_ws, size_t ws_size,
                              hipStream_t stream) {
    const float* x      = (const float*)d_in[0];
    const int*   eidx   = (const int*)  d_in[1];
    const float* attr   = (const float*)d_in[2];
    const float* w1     = (const float*)d_in[3];   // [1,20,32]
    const float* b1     = (const float*)d_in[4];
    const float* w2     = (const float*)d_in[5];   // [2,32,64]
    const float* b2     = (const float*)d_in[6];
    const float* w3     = (const float*)d_in[7];   // [2,64,2]
    const float* b3     = (const float*)d_in[8];
    const float* gate_w = (const float*)d_in[9];   // [2,1]
    const float* gate_b = (const float*)d_in[10];  // [1]
    float* out = (float*)d_out;

    const int N = in_sizes[0] / 20;
    const int E = in_sizes[2];
    const int* src = eidx;
    const int* dst = eidx + E;

    // workspace carving (256B aligned)
    char*  ws  = (char*)d_ws;
    size_t off = 0;
    auto carve = [&](size_t bytes) {
        void* p = ws + off;
        off += (bytes + 255) & ~(size_t)255;
        return p;
    };
    float* deg = (float*)carve((size_t)N * 4);            // becomes dis in place
    float* lw  = (float*)carve((size_t)E * 4);
    float* h1  = (float*)carve((size_t)N * 32 * 4);
    float* t1  = (float*)carve((size_t)N * 32 * 4);
    float* h2  = (float*)carve((size_t)N * 64 * 4);
    float* t2  = (float*)carve((size_t)N * 64 * 4);
    float* h3  = (float*)carve((size_t)N * 2 * 4);
    (void)ws_size;

    const int TB = 256;
    const int n_tiles = (N + 15) / 16;
    const int gemm_blocks = (n_tiles + 7) / 8;

    // 1) Laplacian edge weights
    hipMemsetAsync(deg, 0, (size_t)N * 4, stream);
    deg_scatter_kernel<<<(E + TB - 1) / TB, TB, 0, stream>>>(src, attr, deg, E);
    dis_kernel<<<(N + TB - 1) / TB, TB, 0, stream>>>(deg, N);
    lw_kernel<<<(E + TB - 1) / TB, TB, 0, stream>>>(src, dst, attr, deg, lw, E);

    // 2) Layer 1: h1 = leaky_relu(x @ W1[0] + b1)      [N,20] -> [N,32]
    cheb_gemm_kernel<20, 32, false, true><<<gemm_blocks, TB, 0, stream>>>(
        x, nullptr, w1, nullptr, b1, h1, n_tiles, N);

    // 3) Layer 2: t1 = prop(h1); h2 = leaky_relu(h1@W2[0] + t1@W2[1] + b2)
    hipMemsetAsync(t1, 0, (size_t)N * 32 * 4, stream);
    prop_kernel<32><<<((size_t)E * 8 + TB - 1) / TB, TB, 0, stream>>>(h1, src, dst, lw, t1, E);
    cheb_gemm_kernel<32, 64, true, true><<<gemm_blocks, TB, 0, stream>>>(
        h1, t1, w2, w2 + 32 * 64, b2, h2, n_tiles, N);

    // 4) Layer 3: t2 = prop(h2); h3 = h2@W3[0] + t2@W3[1] + b3
    hipMemsetAsync(t2, 0, (size_t)N * 64 * 4, stream);
    prop_kernel<64><<<((size_t)E * 16 + TB - 1) / TB, TB, 0, stream>>>(h2, src, dst, lw, t2, E);
    cheb_gemm_kernel<64, 2, true, false><<<gemm_blocks, TB, 0, stream>>>(
        h2, t2, w3, w3 + 64 * 2, b3, h3, n_tiles, N);

    // 5) Attention pool + log_softmax -> [1,2]
    finalize_kernel<<<1, 512, 0, stream>>>(h3, gate_w, gate_b, out, N);
}